// HaloAttention_40793599377774
// MI455X (gfx1250) — compile-verified
//
#include <hip/hip_runtime.h>
#include <math.h>

// ---------------------------------------------------------------------------
// HaloAttention (Swin-v2 style, halo'd K/V at half resolution) for gfx1250.
// All GEMMs use V_WMMA_F32_16X16X4_F32 (fp32 in / fp32 accumulate) to match
// the fp32 reference numerics while staying near the HBM roofline.
// Tile staging uses GLOBAL_LOAD_ASYNC_TO_LDS_B128 when the builtin exists.
// ---------------------------------------------------------------------------

typedef float v2f __attribute__((ext_vector_type(2)));
typedef float v8f __attribute__((ext_vector_type(8)));
typedef int   v4i __attribute__((ext_vector_type(4)));

#define WMMA_F32(a, b, c) \
  __builtin_amdgcn_wmma_f32_16x16x4_f32(false, (a), false, (b), (short)0, (c), false, false)

#if defined(__has_builtin)
#if __has_builtin(__builtin_amdgcn_global_load_async_to_lds_b128)
#define HAVE_ASYNC_LDS 1
#endif
#endif
#ifndef HAVE_ASYNC_LDS
#define HAVE_ASYNC_LDS 0
#endif

#if HAVE_ASYNC_LDS
typedef v4i __attribute__((address_space(1)))* gbl_v4i_p;   // global (AS1)
typedef v4i __attribute__((address_space(3)))* lds_v4i_p;   // LDS (AS3)
__device__ __forceinline__ void async_cp16(const float* g, float* l) {
  // per-lane 16B copy, global -> LDS, tracked by ASYNCcnt
  __builtin_amdgcn_global_load_async_to_lds_b128(
      (gbl_v4i_p)(void*)const_cast<float*>(g), (lds_v4i_p)l, 0, 0);
}
__device__ __forceinline__ void async_wait0() {
#if __has_builtin(__builtin_amdgcn_s_wait_asynccnt)
  __builtin_amdgcn_s_wait_asynccnt(0);
#else
  asm volatile("s_wait_asynccnt 0x0" ::: "memory");
#endif
}
#endif

static constexpr int Bb     = 4;
static constexpr int Hh     = 224;
static constexpr int Ww     = 224;
static constexpr int HEADS  = 8;
static constexpr int HH     = 112;                // half resolution
static constexpr int NWIN   = 28 * 28;            // 784 windows
static constexpr int M_FULL = Bb * Hh * Ww;       // 200704 tokens (full res)
static constexpr int P_HALF = Bb * HH * HH;       // 50176 pixels (half res)

// LDS strides (floats): chosen so every 16B staged chunk is 16B-aligned and
// WMMA frag reads are bank-conflict-free across the whole wave.
static constexpr int ASTR = 36;    // A tile row stride (144B)
static constexpr int BSTR = 136;   // B tile row stride (544B)

// ---------------------------------------------------------------------------
// Generic fp32 WMMA GEMM: C[M,N] = A[M,K] * B[K,N] (+epilogue)
// Block tile 128x128, K-chunk 32.  8 waves as 4(M) x 2(N); each wave owns a
// 32x64 subtile = 2x4 tiles of 16x16 -> 8 WMMAs per 6 LDS frag loads.
// MODE 0: qkv epilogue (cols<256 -> q (+q_bias), cols>=256 -> kv_raw).
// MODE 1: out = val + bias (proj).
// ---------------------------------------------------------------------------
template <int MODE>
__global__ __launch_bounds__(256) void gemm_wmma_f32(
    const float* __restrict__ A, const float* __restrict__ Bm,
    const float* __restrict__ bias0, float* __restrict__ out0,
    float* __restrict__ out1, int M, int N, int K) {
  __shared__ float As[128 * ASTR];   // 18.0 KB
  __shared__ float Bs[32 * BSTR];    // 17.0 KB

  const int t    = threadIdx.x;
  const int m0   = blockIdx.y * 128;
  const int n0   = blockIdx.x * 128;
  const int lane = t & 31;
  const int wave = t >> 5;
  const int half = lane >> 4;
  const int lr   = lane & 15;
  const int rm   = (wave & 3) * 32;   // wave M offset in block
  const int cn   = (wave >> 2) * 64;  // wave N offset in block

  v8f acc[2][4] = {};

  for (int kb = 0; kb < K; kb += 32) {
    // ---- stage A (128x32) and B (32x128) ----
#if HAVE_ASYNC_LDS
#pragma unroll
    for (int i = 0; i < 4; ++i) {
      int f = t + i * 256;              // 0..1023 -> 16B chunks of A tile
      int row = f >> 3, col = (f & 7) << 2;
      async_cp16(&A[(size_t)(m0 + row) * K + kb + col], &As[row * ASTR + col]);
    }
#pragma unroll
    for (int i = 0; i < 4; ++i) {
      int f = t + i * 256;              // 0..1023 -> 16B chunks of B tile
      int row = f >> 5, col = (f & 31) << 2;
      async_cp16(&Bm[(size_t)(kb + row) * N + n0 + col], &Bs[row * BSTR + col]);
    }
    async_wait0();
#else
#pragma unroll
    for (int i = 0; i < 4; ++i) {
      int f = t + i * 256;
      int row = f >> 3, col = (f & 7) << 2;
      float4 vA = *(const float4*)(&A[(size_t)(m0 + row) * K + kb + col]);
      *(float4*)(&As[row * ASTR + col]) = vA;
    }
#pragma unroll
    for (int i = 0; i < 4; ++i) {
      int f = t + i * 256;
      int row = f >> 5, col = (f & 31) << 2;
      float4 vB = *(const float4*)(&Bm[(size_t)(kb + row) * N + n0 + col]);
      *(float4*)(&Bs[row * BSTR + col]) = vB;
    }
#endif
    __syncthreads();

#pragma unroll
    for (int kk = 0; kk < 32; kk += 4) {
      const int ks = kk + 2 * half;
      v2f a0, a1;
      a0[0] = As[(rm + lr) * ASTR + ks];      a0[1] = As[(rm + lr) * ASTR + ks + 1];
      a1[0] = As[(rm + 16 + lr) * ASTR + ks]; a1[1] = As[(rm + 16 + lr) * ASTR + ks + 1];
#pragma unroll
      for (int j = 0; j < 4; ++j) {
        v2f b;
        b[0] = Bs[ks * BSTR + cn + j * 16 + lr];
        b[1] = Bs[(ks + 1) * BSTR + cn + j * 16 + lr];
        acc[0][j] = WMMA_F32(a0, b, acc[0][j]);
        acc[1][j] = WMMA_F32(a1, b, acc[1][j]);
      }
    }
    __syncthreads();
  }

  // Epilogue.  C/D layout: lanes 0-15 -> M=r, lanes 16-31 -> M=r+8; N = lr.
#pragma unroll
  for (int ti = 0; ti < 2; ++ti) {
#pragma unroll
    for (int tj = 0; tj < 4; ++tj) {
#pragma unroll
      for (int r = 0; r < 8; ++r) {
        int m = m0 + rm + ti * 16 + r + half * 8;
        int n = n0 + cn + tj * 16 + lr;
        float val = acc[ti][tj][r];
        if constexpr (MODE == 0) {
          if (n < 256)
            out0[(size_t)m * 256 + n] = val + bias0[n];       // q + q_bias
          else
            out1[(size_t)m * 512 + (n - 256)] = val;          // kv (raw)
        } else {
          out0[(size_t)m * N + n] = val + bias0[n];           // proj + bias
        }
      }
    }
  }
}

// ---------------------------------------------------------------------------
// 3x3 depthwise conv, stride 2, SAME (pad_low=0, pad_high=1): 224 -> 112
// ---------------------------------------------------------------------------
__global__ __launch_bounds__(256) void dwconv_k(
    const float* __restrict__ kv_raw, const float* __restrict__ kern,
    float* __restrict__ kv_conv) {
  int idx = blockIdx.x * 256 + threadIdx.x;
  if (idx >= P_HALF * 512) return;
  int c  = idx & 511;
  int pix = idx >> 9;
  int px = pix % HH;
  int py = (pix / HH) % HH;
  int b  = pix / (HH * HH);
  float acc = 0.f;
#pragma unroll
  for (int dy = 0; dy < 3; ++dy) {
    int iy = py * 2 + dy;
    if (iy >= Hh) continue;
#pragma unroll
    for (int dx = 0; dx < 3; ++dx) {
      int ix = px * 2 + dx;
      if (ix >= Ww) continue;
      acc += kv_raw[(((size_t)b * Hh + iy) * Ww + ix) * 512 + c] *
             kern[(dy * 3 + dx) * 512 + c];
    }
  }
  kv_conv[idx] = acc;
}

// ---------------------------------------------------------------------------
// BatchNorm statistics: one block per channel -> mu and rstd
// ---------------------------------------------------------------------------
__global__ __launch_bounds__(256) void bnstats_k(
    const float* __restrict__ kv_conv, float* __restrict__ mu_rstd) {
  __shared__ float s1[256], s2[256];
  int c = blockIdx.x;
  float a = 0.f, b2 = 0.f;
  for (int p = threadIdx.x; p < P_HALF; p += 256) {
    float vv = kv_conv[(size_t)p * 512 + c];
    a += vv; b2 += vv * vv;
  }
  s1[threadIdx.x] = a; s2[threadIdx.x] = b2;
  __syncthreads();
  for (int s = 128; s > 0; s >>= 1) {
    if (threadIdx.x < s) {
      s1[threadIdx.x] += s1[threadIdx.x + s];
      s2[threadIdx.x] += s2[threadIdx.x + s];
    }
    __syncthreads();
  }
  if (threadIdx.x == 0) {
    float mean = s1[0] / (float)P_HALF;
    float var  = s2[0] / (float)P_HALF - mean * mean;
    mu_rstd[c]       = mean;
    mu_rstd[512 + c] = rsqrtf(var + 1.001e-5f);
  }
}

// ---------------------------------------------------------------------------
// BN apply + v_bias + per-(pixel,head) L2-normalize of k
// ---------------------------------------------------------------------------
__global__ __launch_bounds__(256) void kv_point_k(
    const float* __restrict__ kv_conv, const float* __restrict__ mu_rstd,
    const float* __restrict__ gamma, const float* __restrict__ beta,
    const float* __restrict__ v_bias, float* __restrict__ k_norm,
    float* __restrict__ v_out) {
  int idx = blockIdx.x * 256 + threadIdx.x;
  if (idx >= P_HALF * HEADS) return;
  int h = idx & 7;
  size_t p = (size_t)(idx >> 3);
  float kb[32];
  float ss = 0.f;
#pragma unroll
  for (int i = 0; i < 32; ++i) {
    int c = h * 32 + i;
    float x = (kv_conv[p * 512 + c] - mu_rstd[c]) * mu_rstd[512 + c] * gamma[c] + beta[c];
    kb[i] = x; ss += x * x;
  }
  float inv = rsqrtf(fmaxf(ss, 0.00394f));
#pragma unroll
  for (int i = 0; i < 32; ++i) k_norm[p * 256 + h * 32 + i] = kb[i] * inv;
#pragma unroll
  for (int i = 0; i < 32; ++i) {
    int c = 256 + h * 32 + i;
    float x = (kv_conv[p * 512 + c] - mu_rstd[c]) * mu_rstd[512 + c] * gamma[c] +
              beta[c] + v_bias[h * 32 + i];
    v_out[p * 256 + h * 32 + i] = x;
  }
}

// ---------------------------------------------------------------------------
// q: per-(pixel,head) cosine-normalize, multiply by logit_scale (in place)
// ---------------------------------------------------------------------------
__global__ __launch_bounds__(256) void q_point_k(float* __restrict__ q,
                                                 const float* __restrict__ scale) {
  int idx = blockIdx.x * 256 + threadIdx.x;
  if (idx >= M_FULL * HEADS) return;
  int h = idx & 7;
  size_t p = (size_t)(idx >> 3);
  float ls = expf(fminf(scale[h], 4.605170185988092f)); // ln(100)
  float buf[32];
  float ss = 0.f;
#pragma unroll
  for (int i = 0; i < 32; ++i) {
    float x = q[p * 256 + h * 32 + i];
    buf[i] = x; ss += x * x;
  }
  float inv = rsqrtf(fmaxf(ss, 0.00394f)) * ls;
#pragma unroll
  for (int i = 0; i < 32; ++i) q[p * 256 + h * 32 + i] = buf[i] * inv;
}

// ---------------------------------------------------------------------------
// Continuous position bias MLP: bias[h][q(64)][k(64)] = 16*sigmoid(MLP(tab))
// ---------------------------------------------------------------------------
__device__ __forceinline__ float cpb_coord(int qi, int ki) {
  float rel = (float)qi - (2.0f * (float)ki - 4.0f);
  float tv = rel * (8.0f / 7.0f);
  float s = (tv > 0.f) ? 1.f : ((tv < 0.f) ? -1.f : 0.f);
  return s * log2f(1.f + fabsf(tv)) * (1.0f / 3.0f); // /log2(8)
}

__global__ __launch_bounds__(64) void cpb_k(
    const float* __restrict__ w1, const float* __restrict__ b1,
    const float* __restrict__ w2, float* __restrict__ biasT) {
  int qidx = blockIdx.x, kidx = threadIdx.x;
  float ry = cpb_coord(qidx >> 3, kidx >> 3);
  float rx = cpb_coord(qidx & 7, kidx & 7);
  float acc[8] = {};
  for (int hid = 0; hid < 512; ++hid) {
    float hv = fmaxf(ry * w1[hid] + rx * w1[512 + hid] + b1[hid], 0.f);
#pragma unroll
    for (int h = 0; h < 8; ++h) acc[h] += hv * w2[hid * 8 + h];
  }
#pragma unroll
  for (int h = 0; h < 8; ++h)
    biasT[((size_t)h * 64 + qidx) * 64 + kidx] = 16.f / (1.f + expf(-acc[h]));
}

// ---------------------------------------------------------------------------
// Fused halo attention.  One block = (batch, window, head).  4 waves.
// S = Q(64x32) . K^T(32x64) via WMMA, +bias +halo-mask, softmax, O = P.V.
// ---------------------------------------------------------------------------
__global__ __launch_bounds__(128) void halo_attn_k(
    const float* __restrict__ q, const float* __restrict__ k_norm,
    const float* __restrict__ v, const float* __restrict__ biasT,
    float* __restrict__ attn_out) {
  __shared__ float q_s[64][33];
  __shared__ float k_s[64][33];
  __shared__ float v_s[64][33];
  __shared__ float s_s[64][65];
  __shared__ float validm[64];   // 0 if in-bounds, -100 if padded halo pos

  const int win = blockIdx.x, head = blockIdx.y, b = blockIdx.z;
  const int wy = win / 28, wx = win % 28;
  const int t = threadIdx.x;

  // Stage q window (8x8 tokens x 32 ch)
  {
    int m = t >> 1, seg = t & 1;
    int py = wy * 8 + (m >> 3), px = wx * 8 + (m & 7);
    const float* src = &q[(((size_t)b * Hh + py) * Ww + px) * 256 + head * 32 + seg * 16];
#pragma unroll
    for (int j = 0; j < 4; ++j) {
      float4 vv = *(const float4*)(src + j * 4);
      q_s[m][seg * 16 + j * 4 + 0] = vv.x; q_s[m][seg * 16 + j * 4 + 1] = vv.y;
      q_s[m][seg * 16 + j * 4 + 2] = vv.z; q_s[m][seg * 16 + j * 4 + 3] = vv.w;
    }
  }
  // Stage k/v halo window (8x8 at half res, offset -2, zero padded)
  {
    int m = t >> 1, seg = t & 1;
    int gy = wy * 4 + (m >> 3) - 2, gx = wx * 4 + (m & 7) - 2;
    bool ok = (gy >= 0 && gy < HH && gx >= 0 && gx < HH);
    if (seg == 0) validm[m] = ok ? 0.f : -100.f;
    size_t base = (((size_t)b * HH + (ok ? gy : 0)) * HH + (ok ? gx : 0)) * 256 +
                  head * 32 + seg * 16;
#pragma unroll
    for (int j = 0; j < 4; ++j) {
      float4 kk4 = ok ? *(const float4*)(&k_norm[base + j * 4]) : make_float4(0, 0, 0, 0);
      float4 vv4 = ok ? *(const float4*)(&v[base + j * 4]) : make_float4(0, 0, 0, 0);
      k_s[m][seg * 16 + j * 4 + 0] = kk4.x; k_s[m][seg * 16 + j * 4 + 1] = kk4.y;
      k_s[m][seg * 16 + j * 4 + 2] = kk4.z; k_s[m][seg * 16 + j * 4 + 3] = kk4.w;
      v_s[m][seg * 16 + j * 4 + 0] = vv4.x; v_s[m][seg * 16 + j * 4 + 1] = vv4.y;
      v_s[m][seg * 16 + j * 4 + 2] = vv4.z; v_s[m][seg * 16 + j * 4 + 3] = vv4.w;
    }
  }
  __syncthreads();

  const int lane = t & 31, wave = t >> 5;
  const int half = lane >> 4, lr = lane & 15;
  const int m0 = wave * 16;  // each wave owns 16 query rows

  // S = Q . K^T  (K-dim = 32 channels)
  v8f sacc[4] = {};
#pragma unroll
  for (int kk = 0; kk < 32; kk += 4) {
    const int ks = kk + 2 * half;
    v2f a; a[0] = q_s[m0 + lr][ks]; a[1] = q_s[m0 + lr][ks + 1];
#pragma unroll
    for (int i = 0; i < 4; ++i) {
      v2f bf; bf[0] = k_s[i * 16 + lr][ks]; bf[1] = k_s[i * 16 + lr][ks + 1];
      sacc[i] = WMMA_F32(a, bf, sacc[i]);
    }
  }
  // scores + CPB bias + halo mask -> LDS
#pragma unroll
  for (int i = 0; i < 4; ++i) {
#pragma unroll
    for (int r = 0; r < 8; ++r) {
      int m = m0 + r + half * 8;
      int col = i * 16 + lr;
      s_s[m][col] = sacc[i][r] + biasT[((size_t)head * 64 + m) * 64 + col] + validm[col];
    }
  }
  __syncthreads();

  // Row softmax over 64 halo keys
  if (t < 64) {
    float mx = -1e30f;
    for (int j = 0; j < 64; ++j) mx = fmaxf(mx, s_s[t][j]);
    float sm = 0.f;
    for (int j = 0; j < 64; ++j) { float e = expf(s_s[t][j] - mx); s_s[t][j] = e; sm += e; }
    float inv = 1.f / sm;
    for (int j = 0; j < 64; ++j) s_s[t][j] *= inv;
  }
  __syncthreads();

  // O = P(64x64) . V(64x32)
  v8f oacc[2] = {};
#pragma unroll
  for (int kk = 0; kk < 64; kk += 4) {
    const int ks = kk + 2 * half;
    v2f a; a[0] = s_s[m0 + lr][ks]; a[1] = s_s[m0 + lr][ks + 1];
#pragma unroll
    for (int i = 0; i < 2; ++i) {
      v2f bf; bf[0] = v_s[ks][i * 16 + lr]; bf[1] = v_s[ks + 1][i * 16 + lr];
      oacc[i] = WMMA_F32(a, bf, oacc[i]);
    }
  }
  // window-reverse scatter store
#pragma unroll
  for (int i = 0; i < 2; ++i) {
#pragma unroll
    for (int r = 0; r < 8; ++r) {
      int m = m0 + r + half * 8;
      int py = wy * 8 + (m >> 3), px = wx * 8 + (m & 7);
      attn_out[(((size_t)b * Hh + py) * Ww + px) * 256 + head * 32 + i * 16 + lr] = oacc[i][r];
    }
  }
}

// ---------------------------------------------------------------------------
extern "C" void kernel_launch(void* const* d_in, const int* in_sizes, int n_in,
                              void* d_out, int out_size, void* d_ws, size_t ws_size,
                              hipStream_t stream) {
  const float* x      = (const float*)d_in[0];
  const float* w_qkv  = (const float*)d_in[1];
  const float* q_bias = (const float*)d_in[2];
  const float* v_bias = (const float*)d_in[3];
  const float* dw_k   = (const float*)d_in[4];
  const float* bn_g   = (const float*)d_in[5];
  const float* bn_b   = (const float*)d_in[6];
  const float* scale  = (const float*)d_in[7];
  const float* cpb_w1 = (const float*)d_in[8];
  const float* cpb_b1 = (const float*)d_in[9];
  const float* cpb_w2 = (const float*)d_in[10];
  const float* proj_w = (const float*)d_in[11];
  const float* proj_b = (const float*)d_in[12];
  float* out = (float*)d_out;
  float* ws  = (float*)d_ws;

  size_t off = 0;
  float* qbuf    = ws + off; off += (size_t)M_FULL * 256;  // q (normalized in place)
  float* kv_conv = ws + off; off += (size_t)P_HALF * 512;
  float* k_norm  = ws + off; off += (size_t)P_HALF * 256;
  float* vbuf    = ws + off; off += (size_t)P_HALF * 256;
  float* biasT   = ws + off; off += 8 * 64 * 64;
  float* mu_rstd = ws + off; off += 1024;
  float* kv_raw  = ws + off; off += (size_t)M_FULL * 512;
  float* attn_o  = kv_raw;  // kv_raw is dead after the conv; reuse region

  // 1) qkv = x @ w_qkv, split into q(+q_bias) and kv_raw
  gemm_wmma_f32<0><<<dim3(768 / 128, M_FULL / 128), 256, 0, stream>>>(
      x, w_qkv, q_bias, qbuf, kv_raw, M_FULL, 768, 256);
  // 2) depthwise conv stride-2
  dwconv_k<<<(P_HALF * 512 + 255) / 256, 256, 0, stream>>>(kv_raw, dw_k, kv_conv);
  // 3) BN statistics
  bnstats_k<<<512, 256, 0, stream>>>(kv_conv, mu_rstd);
  // 4) BN apply + v_bias, k cosine-normalize
  kv_point_k<<<(P_HALF * HEADS + 255) / 256, 256, 0, stream>>>(
      kv_conv, mu_rstd, bn_g, bn_b, v_bias, k_norm, vbuf);
  // 5) q cosine-normalize * logit_scale
  q_point_k<<<(M_FULL * HEADS + 255) / 256, 256, 0, stream>>>(qbuf, scale);
  // 6) CPB bias table
  cpb_k<<<64, 64, 0, stream>>>(cpb_w1, cpb_b1, cpb_w2, biasT);
  // 7) fused halo attention
  halo_attn_k<<<dim3(NWIN, HEADS, Bb), 128, 0, stream>>>(qbuf, k_norm, vbuf, biasT, attn_o);
  // 8) projection
  gemm_wmma_f32<1><<<dim3(256 / 128, M_FULL / 128), 256, 0, stream>>>(
      attn_o, proj_w, proj_b, out, nullptr, M_FULL, 256, 256);

  (void)in_sizes; (void)n_in; (void)out_size; (void)ws_size;
}